// FIRResample2d_21251498181272
// MI455X (gfx1250) — compile-verified
//
#include <hip/hip_runtime.h>
#include <stdint.h>

// upfirdn2d: up=2, down=1, pad=(2,1), 4x4 kernel, fp32.
// In : (16,128,128,128) NCHW  -> Out: (16,128,256,256) NCHW.
// Memory-bound (~671MB @ 23.3TB/s ~ 29us); use async global->LDS staging
// (ASYNCcnt path) + non-temporal b128 output stores to keep the 128MB input
// resident in the 192MB L2 while streaming the 512MB output past it.

#define IN_H 128
#define IN_W 128
#define OUT_H (2 * IN_H)
#define OUT_W (2 * IN_W)
#define TILE 32                   // input tile core (-> 64x64 output tile)
#define HALO_DIM (TILE + 2)       // 34: one halo ring
#define LDS_STRIDE (HALO_DIM + 1) // 35: pad to dodge bank conflicts
#define NTHREADS 256

typedef float v4f __attribute__((ext_vector_type(4)));
typedef __attribute__((address_space(1))) int as1_int; // HIP prints as "__device__ int"
typedef __attribute__((address_space(3))) int as3_int; // LDS

// ---- CDNA5 async global->LDS load (tracked by ASYNCcnt) ------------------
#if defined(__has_builtin) && __has_builtin(__builtin_amdgcn_global_load_async_to_lds_b32)
#define ASYNC_G2L_B32(gptr, lptr)                                              \
  __builtin_amdgcn_global_load_async_to_lds_b32(                               \
      (as1_int*)(uintptr_t)(gptr), (as3_int*)(uintptr_t)(lptr), 0, 0)
#else
// GV form: vdst = LDS byte address, vaddr = 64-bit global address, saddr off
#define ASYNC_G2L_B32(gptr, lptr)                                              \
  asm volatile("global_load_async_to_lds_b32 %0, %1, off"                      \
               :                                                               \
               : "v"((uint32_t)(uintptr_t)(lptr)),                             \
                 "v"((uint64_t)(uintptr_t)(gptr))                              \
               : "memory")
#endif

#if defined(__has_builtin) && __has_builtin(__builtin_amdgcn_s_wait_asynccnt)
#define WAIT_ASYNC0() __builtin_amdgcn_s_wait_asynccnt(0)
#else
#define WAIT_ASYNC0() asm volatile("s_wait_asynccnt 0" ::: "memory")
#endif
// --------------------------------------------------------------------------

__global__ __launch_bounds__(NTHREADS) void
upfirdn2x_kernel(const float* __restrict__ x, const float* __restrict__ k2,
                 float* __restrict__ out) {
  __shared__ float tile[HALO_DIM * LDS_STRIDE];

  const int tid   = threadIdx.x;
  const int tX    = blockIdx.x;  // 0..3 tile col
  const int tY    = blockIdx.y;  // 0..3 tile row
  const int bc    = blockIdx.z;  // 0..2047 fused batch*channel
  const int baseY = tY * TILE;
  const int baseX = tX * TILE;

  // Zero whole LDS tile: image-edge halo lanes issue no load and must read 0.
#pragma unroll
  for (int i = tid; i < HALO_DIM * LDS_STRIDE; i += NTHREADS) tile[i] = 0.0f;
  __syncthreads();

  // Async-gather the 34x34 haloed input window into LDS.
  const float* src = x + (size_t)bc * (IN_H * IN_W);
  for (int i = tid; i < HALO_DIM * HALO_DIM; i += NTHREADS) {
    const int r  = i / HALO_DIM;
    const int c  = i - r * HALO_DIM;
    const int gy = baseY + r - 1;
    const int gx = baseX + c - 1;
    if ((unsigned)gy < (unsigned)IN_H && (unsigned)gx < (unsigned)IN_W) {
      ASYNC_G2L_B32(src + (size_t)gy * IN_W + gx, &tile[r * LDS_STRIDE + c]);
    }
  }
  WAIT_ASYNC0();
  __syncthreads();

  // Flipped kernel (true convolution). Uniform reads -> scalar loads.
  float kf[4][4];
#pragma unroll
  for (int i = 0; i < 4; ++i)
#pragma unroll
    for (int j = 0; j < 4; ++j) kf[i][j] = k2[(3 - i) * 4 + (3 - j)];

  // Each thread: 2x2 input core block -> 4x4 output block.
  const int r0 = (tid >> 4) << 1; // 0..30 (even)
  const int c0 = (tid & 15) << 1; // 0..30 (even)

  float s[4][4]; // LDS rows r0..r0+3 <-> input rows (baseY+r0-1)..(+2)
#pragma unroll
  for (int i = 0; i < 4; ++i)
#pragma unroll
    for (int j = 0; j < 4; ++j)
      s[i][j] = tile[(r0 + i) * LDS_STRIDE + (c0 + j)];

  const int oy0 = (baseY + r0) * 2;
  const int ox0 = (baseX + c0) * 2; // multiple of 4 -> 16B-aligned stores
  float* dst = out + ((size_t)bc * OUT_H + oy0) * OUT_W + ox0;

#pragma unroll
  for (int k = 0; k < 4; ++k) {
    const int dy = k >> 1, py = k & 1;
    v4f v;
#pragma unroll
    for (int l = 0; l < 4; ++l) {
      const int dx = l >> 1, px = l & 1;
      float acc = 0.0f;
#pragma unroll
      for (int a = 0; a < 2; ++a) {
        const int ty = py ? (2 * a + 1) : (2 * a); // contributing tap (vert)
        const int si = dy + ((ty + 1) >> 1);
#pragma unroll
        for (int b = 0; b < 2; ++b) {
          const int tx = px ? (2 * b + 1) : (2 * b); // contributing tap (horz)
          const int sj = dx + ((tx + 1) >> 1);
          acc += kf[ty][tx] * s[si][sj];
        }
      }
      v[l] = acc;
    }
    // Non-temporal: don't let the 512MB output stream evict the input from L2.
    __builtin_nontemporal_store(v, (v4f*)(dst + (size_t)k * OUT_W));
  }
}

extern "C" void kernel_launch(void* const* d_in, const int* in_sizes, int n_in,
                              void* d_out, int out_size, void* d_ws,
                              size_t ws_size, hipStream_t stream) {
  (void)in_sizes; (void)n_in; (void)d_ws; (void)ws_size; (void)out_size;
  const float* x  = (const float*)d_in[0];
  const float* k2 = (const float*)d_in[1];
  float* out      = (float*)d_out;

  dim3 grid(IN_W / TILE, IN_H / TILE, 16 * 128); // (4,4,2048)
  dim3 block(NTHREADS);
  upfirdn2x_kernel<<<grid, block, 0, stream>>>(x, k2, out);
}